// BlurNet_7198365188390
// MI455X (gfx1250) — compile-verified
//
#include <hip/hip_runtime.h>

// BlurNet bilinear warp + offsets emit, tuned for MI455X (gfx1250, wave32).
// Bandwidth-bound: ~79 MB total traffic -> ~3.4us floor at 23.3 TB/s.
// No matrix structure -> WMMA not applicable; optimize for coalesced wave32
// stores, WGP$-resident gathers, and gfx1250 WGP-scope global_prefetch.

namespace {
constexpr int BATCH = 8;
constexpr int CH    = 3;
constexpr int HDIM  = 256;
constexpr int WDIM  = 256;
constexpr int HW    = HDIM * WDIM;
constexpr int NOFF  = 15;
} // namespace

// WGP-scope prefetch (scope field omitted -> scope 0 = WGP, which per the
// CDNA5 ISA pulls the line into ALL cache levels, including WGP$).
// __builtin_prefetch can only reach SE/DEV/SYS scope, which skip the WGP$.
__device__ __forceinline__ void prefetch_wgp(const void* p) {
    asm volatile("global_prefetch_b8 %0, off" :: "v"(p));
}

__global__ __launch_bounds__(256)
void blurnet_warp_kernel(const float* __restrict__ fake_S,   // (8,3,256,256)
                         const float* __restrict__ blurmap,  // (8,2,256,256)
                         const float* __restrict__ weight,   // (1)
                         const float* __restrict__ bias,     // (1)
                         float* __restrict__ out)            // fake_B | offsets
{
    const int w  = threadIdx.x;            // 0..255 -> coalesced along W
    const int bh = blockIdx.x;             // b*256 + h
    const int b  = bh >> 8;
    const int h  = bh & (HDIM - 1);
    const int hw = (h << 8) | w;

    const float wgt = weight[0];
    const float bs  = bias[0];

    // off_S = 20 * blurmap  (channel 0 -> dy scale, channel 1 -> dx scale)
    const float m0 = 20.0f * blurmap[((b << 1) + 0) * HW + hw];
    const float m1 = 20.0f * blurmap[((b << 1) + 1) * HW + hw];

    const float* __restrict__ img = fake_S + (size_t)b * CH * HW;

    // Warm the center row +/-1 of each channel into WGP$: every small-|k|
    // tap (incl. the exact-center k=7 tap) hits these rows, and the rest of
    // the +-20px band follows into WGP$ on first gather.
#pragma unroll
    for (int c = 0; c < CH; ++c) {
        const float* pc = img + c * HW + hw;
        prefetch_wgp(pc - WDIM);
        prefetch_wgp(pc);
        prefetch_wgp(pc + WDIM);
    }

    float acc0 = 0.0f, acc1 = 0.0f, acc2 = 0.0f;

    float* __restrict__ out_fb  = out;                           // 8*3*HW
    float* __restrict__ out_off = out + (size_t)BATCH * CH * HW; // 8*30*HW

#pragma unroll
    for (int k = 0; k < NOFF; ++k) {
        // coef_k = (7-k)/7  (k=0..6 forward leg, k=7 zero, k=8..14 backward leg)
        const float coef = (float)(7 - k) / 7.0f;
        const float dy = coef * m0;
        const float dx = coef * m1;

        // offsets_flat plane j = 2k (dy) and 2k+1 (dx); coalesced per wave
        out_off[(size_t)(b * 30 + 2 * k + 0) * HW + hw] = dy;
        out_off[(size_t)(b * 30 + 2 * k + 1) * HW + hw] = dx;

        const float y   = (float)h + dy;
        const float x   = (float)w + dx;
        const float y0f = floorf(y);
        const float x0f = floorf(x);
        const float ty  = y - y0f;
        const float tx  = x - x0f;
        const int   y0  = (int)y0f;
        const int   x0  = (int)x0f;
        const int   y1  = y0 + 1;
        const int   x1  = x0 + 1;

        // per-corner validity (zero padding outside the image)
        const bool vy0 = (unsigned)y0 < (unsigned)HDIM;
        const bool vy1 = (unsigned)y1 < (unsigned)HDIM;
        const bool vx0 = (unsigned)x0 < (unsigned)WDIM;
        const bool vx1 = (unsigned)x1 < (unsigned)WDIM;
        const float f00 = (vy0 && vx0) ? 1.0f : 0.0f;
        const float f01 = (vy0 && vx1) ? 1.0f : 0.0f;
        const float f10 = (vy1 && vx0) ? 1.0f : 0.0f;
        const float f11 = (vy1 && vx1) ? 1.0f : 0.0f;

        // clamped gather indices (shared across the 3 channels)
        const int y0c = min(max(y0, 0), HDIM - 1);
        const int y1c = min(max(y1, 0), HDIM - 1);
        const int x0c = min(max(x0, 0), WDIM - 1);
        const int x1c = min(max(x1, 0), WDIM - 1);
        const int i00 = (y0c << 8) | x0c;
        const int i01 = (y0c << 8) | x1c;
        const int i10 = (y1c << 8) | x0c;
        const int i11 = (y1c << 8) | x1c;

        const float omty = 1.0f - ty;
        const float omtx = 1.0f - tx;
        const float w00 = omty * omtx;
        const float w01 = omty * tx;
        const float w10 = ty * omtx;
        const float w11 = ty * tx;

        const float* __restrict__ p0 = img;
        const float* __restrict__ p1 = img + HW;
        const float* __restrict__ p2 = img + 2 * HW;

        acc0 += (p0[i00] * f00) * w00 + (p0[i01] * f01) * w01 +
                (p0[i10] * f10) * w10 + (p0[i11] * f11) * w11;
        acc1 += (p1[i00] * f00) * w00 + (p1[i01] * f01) * w01 +
                (p1[i10] * f10) * w10 + (p1[i11] * f11) * w11;
        acc2 += (p2[i00] * f00) * w00 + (p2[i01] * f01) * w01 +
                (p2[i10] * f10) * w10 + (p2[i11] * f11) * w11;
    }

    // fake_B = weight * mean(sampled over 15) + bias
    out_fb[(size_t)(b * CH + 0) * HW + hw] = wgt * (acc0 / 15.0f) + bs;
    out_fb[(size_t)(b * CH + 1) * HW + hw] = wgt * (acc1 / 15.0f) + bs;
    out_fb[(size_t)(b * CH + 2) * HW + hw] = wgt * (acc2 / 15.0f) + bs;
}

extern "C" void kernel_launch(void* const* d_in, const int* in_sizes, int n_in,
                              void* d_out, int out_size, void* d_ws, size_t ws_size,
                              hipStream_t stream) {
    (void)in_sizes; (void)n_in; (void)out_size; (void)d_ws; (void)ws_size;
    const float* fake_S  = (const float*)d_in[0];
    const float* blurmap = (const float*)d_in[1];
    const float* weight  = (const float*)d_in[2];
    const float* bias    = (const float*)d_in[3];
    float* out = (float*)d_out;

    dim3 grid(BATCH * HDIM);   // 2048 blocks: one (b,h) row each
    dim3 block(WDIM);          // 256 threads = 8 wave32 per block
    hipLaunchKernelGGL(blurnet_warp_kernel, grid, block, 0, stream,
                       fake_S, blurmap, weight, bias, out);
}